// MultiHeadAttention_68891275428448
// MI455X (gfx1250) — compile-verified
//
#include <hip/hip_runtime.h>
#include <math.h>

typedef __attribute__((ext_vector_type(2))) float v2f;
typedef __attribute__((ext_vector_type(4))) float v4f;
typedef __attribute__((ext_vector_type(8))) float v8f;

#define B_  2
#define C_  2048
#define E_  1024
#define H_  16
#define D_  64
#define HD_ 1024
#define BC_ 4096

static __device__ __forceinline__ v8f vzero8() {
  v8f z = {0.f, 0.f, 0.f, 0.f, 0.f, 0.f, 0.f, 0.f};
  return z;
}

static __device__ __forceinline__ v8f wmma_f32(v2f a, v2f b, v8f c) {
  // D = A(16x4,f32) x B(4x16,f32) + C(16x16,f32)
  return __builtin_amdgcn_wmma_f32_16x16x4_f32(false, a, false, b, (short)0, c,
                                               false, false);
}

// ---------------------------------------------------------------------------
// Kernel 1: fused QKV projection. Each wave computes a 16x64 tile (exactly one
// head's width) of the concatenated [Q|K|V] output (width 3*1024): 4
// accumulators share each A fragment, so 4 WMMAs per LDS A-read.
// x-slab staged via LDS. Output written directly in (B,H,C,D) layout.
// ---------------------------------------------------------------------------
__global__ __launch_bounds__(256) void qkv_kernel(
    const float* __restrict__ x, const float* __restrict__ Wq,
    const float* __restrict__ Wk, const float* __restrict__ Wv,
    float* __restrict__ Qo, float* __restrict__ Ko, float* __restrict__ Vo) {
  __shared__ float xs[16 * 260];  // 16 rows x 256 k-chunk, stride 260 (pad 4)
  const int tid = threadIdx.x;
  const int wave = tid >> 5, lane = tid & 31, h = lane >> 4, m = lane & 15;
  const int row0 = blockIdx.x * 16;
  const int n0g = (blockIdx.y * 8 + wave) * 64;  // column in [0, 3072), 64-wide
  const int mat = n0g >> 10;                     // 0=Q, 1=K, 2=V
  const int jin = n0g & 1023;                    // multiple of 64 -> one head
  const float* Wm = (mat == 0) ? Wq : ((mat == 1) ? Wk : Wv);
  float* dst = (mat == 0) ? Qo : ((mat == 1) ? Ko : Vo);

  v8f acc0 = vzero8(), acc1 = vzero8(), acc2 = vzero8(), acc3 = vzero8();
  for (int kc = 0; kc < E_; kc += 256) {
    __syncthreads();
#pragma unroll
    for (int it = 0; it < 4; ++it) {
      int q = tid + it * 256;
      int r = q >> 6, c4 = (q & 63) << 2;
      v4f v = *(const v4f*)&x[(size_t)(row0 + r) * E_ + kc + c4];
      *(v4f*)&xs[r * 260 + c4] = v;
    }
    __syncthreads();
#pragma unroll 2
    for (int k = 0; k < 256; k += 4) {
      v2f a = *(const v2f*)&xs[m * 260 + k + 2 * h];
      const int krow = kc + k + 2 * h;
      const float* w0 = &Wm[(size_t)krow * HD_ + jin + m];
      v2f b;
      b.x = w0[0 * 16];  b.y = w0[HD_ + 0 * 16];  acc0 = wmma_f32(a, b, acc0);
      b.x = w0[1 * 16];  b.y = w0[HD_ + 1 * 16];  acc1 = wmma_f32(a, b, acc1);
      b.x = w0[2 * 16];  b.y = w0[HD_ + 2 * 16];  acc2 = wmma_f32(a, b, acc2);
      b.x = w0[3 * 16];  b.y = w0[HD_ + 3 * 16];  acc3 = wmma_f32(a, b, acc3);
    }
  }
  const int head = jin >> 6;
#pragma unroll
  for (int r = 0; r < 8; ++r) {
    int t = row0 + 8 * h + r;         // token index in [0, B*C)
    int bb = t >> 11, cc = t & 2047;  // C_ == 2048
    float* o = &dst[(((size_t)bb * H_ + head) * C_ + cc) * D_ + m];
    o[0 * 16] = acc0[r];
    o[1 * 16] = acc1[r];
    o[2 * 16] = acc2[r];
    o[3 * 16] = acc3[r];
  }
}

// ---------------------------------------------------------------------------
// Kernel 2: RoPE applied in place to Q and K, (B,H,C,D) layout.
// ---------------------------------------------------------------------------
__global__ __launch_bounds__(256) void rope_kernel(float* __restrict__ Q,
                                                   float* __restrict__ K) {
  const int idx = blockIdx.x * 256 + threadIdx.x;  // over B*H*C*(D/2)
  float* P = (blockIdx.y == 0) ? Q : K;
  const int i = idx & 31;            // pair index in head dim
  const int c = (idx >> 5) & 2047;   // position
  const int bh = idx >> 16;          // (b,h) flat, C*32 = 65536 pairs each
  const float inv_freq = __expf(-(2.0f * (float)i / 64.0f) * 9.210340371976184f);
  const float ang = (float)c * inv_freq;
  float sn, cs;
  sincosf(ang, &sn, &cs);
  const size_t off = ((size_t)bh * C_ + c) * D_ + 2 * i;
  const float e = P[off], o = P[off + 1];
  P[off] = e * cs - o * sn;
  P[off + 1] = e * sn + o * cs;
}

// ---------------------------------------------------------------------------
// Kernel 3: causal flash attention. One wave per 16-query tile.
// S = Q K^T via 16 f32 WMMAs; online softmax with shfl_xor row reductions;
// P transposed through per-wave LDS; O += P V via 16 f32 WMMAs.
// ---------------------------------------------------------------------------
__global__ __launch_bounds__(256) void attn_kernel(
    const float* __restrict__ Q, const float* __restrict__ K,
    const float* __restrict__ V, float* __restrict__ hid) {
  __shared__ float pbuf[8][16 * 17];  // per-wave P scratch, padded stride 17
  const int tid = threadIdx.x;
  const int wave = tid >> 5, lane = tid & 31, h = lane >> 4, m = lane & 15;
  const int bh = blockIdx.y;
  const int bb = bh >> 4, head = bh & 15;
  const int qt = blockIdx.x * 8 + wave;
  const int q0 = qt * 16;
  const size_t base = (size_t)bh * C_ * D_;
  float* P = pbuf[wave];

  // Q tile as 16 A-fragments (K-dim = D = 64, 16 steps of 4)
  v2f aq[16];
#pragma unroll
  for (int s = 0; s < 16; ++s)
    aq[s] = *(const v2f*)&Q[base + (size_t)(q0 + m) * D_ + 4 * s + 2 * h];

  v8f o0 = vzero8(), o1 = vzero8(), o2 = vzero8(), o3 = vzero8();
  float mrow[8], lrow[8];
#pragma unroll
  for (int r = 0; r < 8; ++r) { mrow[r] = -INFINITY; lrow[r] = 0.f; }

  for (int kt = 0; kt <= qt; ++kt) {
    const int k0 = kt * 16;
    // S = Q K^T  (M=q, N=k-token, reduce over head dim)
    v8f sc = vzero8();
#pragma unroll
    for (int s = 0; s < 16; ++s) {
      v2f bk = *(const v2f*)&K[base + (size_t)(k0 + m) * D_ + 4 * s + 2 * h];
      sc = wmma_f32(aq[s], bk, sc);
    }
    // scale, causal mask, online softmax
    float corr[8];
#pragma unroll
    for (int r = 0; r < 8; ++r) {
      float sv = sc[r] * 0.125f;  // 1/sqrt(64)
      if (kt == qt && (8 * h + r) < m) sv = -INFINITY;
      float vmax = sv;
#pragma unroll
      for (int off = 1; off < 16; off <<= 1)
        vmax = fmaxf(vmax, __shfl_xor(vmax, off, 32));
      float mnew = fmaxf(mrow[r], vmax);
      float cr = __expf(mrow[r] - mnew);
      float p = __expf(sv - mnew);
      float rs = p;
#pragma unroll
      for (int off = 1; off < 16; off <<= 1) rs += __shfl_xor(rs, off, 32);
      lrow[r] = lrow[r] * cr + rs;
      mrow[r] = mnew;
      corr[r] = cr;
      P[(r + 8 * h) * 17 + m] = p;  // store C-layout -> [q_local][k_local]
    }
#pragma unroll
    for (int r = 0; r < 8; ++r) {
      o0[r] *= corr[r]; o1[r] *= corr[r]; o2[r] *= corr[r]; o3[r] *= corr[r];
    }
    // make P stores visible to our own A-layout reads (same-wave LDS RAW)
    asm volatile("s_wait_dscnt 0x0" ::: "memory");
    // O += P V  (M=q, N=d, reduce over 16 k-tokens: 4 steps of 4)
#pragma unroll
    for (int s = 0; s < 4; ++s) {
      const int kl = 4 * s + 2 * h;
      v2f ap;
      ap.x = P[m * 17 + kl];
      ap.y = P[m * 17 + kl + 1];
      const float* v0 = &V[base + (size_t)(k0 + kl) * D_ + m];
      const float* v1 = v0 + D_;
      v2f bv;
      bv.x = v0[0 * 16]; bv.y = v1[0 * 16];  o0 = wmma_f32(ap, bv, o0);
      bv.x = v0[1 * 16]; bv.y = v1[1 * 16];  o1 = wmma_f32(ap, bv, o1);
      bv.x = v0[2 * 16]; bv.y = v1[2 * 16];  o2 = wmma_f32(ap, bv, o2);
      bv.x = v0[3 * 16]; bv.y = v1[3 * 16];  o3 = wmma_f32(ap, bv, o3);
    }
  }
  // normalize and write hidden in (B, C, H*D) layout
#pragma unroll
  for (int r = 0; r < 8; ++r) {
    const float inv = 1.0f / lrow[r];
    float* o = &hid[((size_t)bb * C_ + q0 + 8 * h + r) * HD_ + head * D_ + m];
    o[0 * 16] = o0[r] * inv;
    o[1 * 16] = o1[r] * inv;
    o[2 * 16] = o2[r] * inv;
    o[3 * 16] = o3[r] * inv;
  }
}

// ---------------------------------------------------------------------------
// Kernel 4: output projection hidden(4096x1024) @ Wo(1024x1024) -> out.
// 16x64 tile per wave, same structure as kernel 1.
// ---------------------------------------------------------------------------
__global__ __launch_bounds__(256) void proj_kernel(
    const float* __restrict__ hid, const float* __restrict__ Wo,
    float* __restrict__ out) {
  __shared__ float xs[16 * 260];
  const int tid = threadIdx.x;
  const int wave = tid >> 5, lane = tid & 31, h = lane >> 4, m = lane & 15;
  const int row0 = blockIdx.x * 16;
  const int n0 = (blockIdx.y * 8 + wave) * 64;

  v8f acc0 = vzero8(), acc1 = vzero8(), acc2 = vzero8(), acc3 = vzero8();
  for (int kc = 0; kc < HD_; kc += 256) {
    __syncthreads();
#pragma unroll
    for (int it = 0; it < 4; ++it) {
      int q = tid + it * 256;
      int r = q >> 6, c4 = (q & 63) << 2;
      v4f v = *(const v4f*)&hid[(size_t)(row0 + r) * HD_ + kc + c4];
      *(v4f*)&xs[r * 260 + c4] = v;
    }
    __syncthreads();
#pragma unroll 2
    for (int k = 0; k < 256; k += 4) {
      v2f a = *(const v2f*)&xs[m * 260 + k + 2 * h];
      const int krow = kc + k + 2 * h;
      const float* w0 = &Wo[(size_t)krow * E_ + n0 + m];
      v2f b;
      b.x = w0[0 * 16];  b.y = w0[E_ + 0 * 16];  acc0 = wmma_f32(a, b, acc0);
      b.x = w0[1 * 16];  b.y = w0[E_ + 1 * 16];  acc1 = wmma_f32(a, b, acc1);
      b.x = w0[2 * 16];  b.y = w0[E_ + 2 * 16];  acc2 = wmma_f32(a, b, acc2);
      b.x = w0[3 * 16];  b.y = w0[E_ + 3 * 16];  acc3 = wmma_f32(a, b, acc3);
    }
  }
#pragma unroll
  for (int r = 0; r < 8; ++r) {
    float* o = &out[(size_t)(row0 + 8 * h + r) * E_ + n0 + m];
    o[0 * 16] = acc0[r];
    o[1 * 16] = acc1[r];
    o[2 * 16] = acc2[r];
    o[3 * 16] = acc3[r];
  }
}

// ---------------------------------------------------------------------------
extern "C" void kernel_launch(void* const* d_in, const int* in_sizes, int n_in,
                              void* d_out, int out_size, void* d_ws,
                              size_t ws_size, hipStream_t stream) {
  (void)in_sizes; (void)n_in; (void)out_size; (void)ws_size;
  const float* x  = (const float*)d_in[0];
  const float* Wq = (const float*)d_in[1];
  const float* Wk = (const float*)d_in[2];
  const float* Wv = (const float*)d_in[3];
  const float* Wo = (const float*)d_in[4];
  float* out = (float*)d_out;

  const size_t BHCD = (size_t)B_ * H_ * C_ * D_;  // 4,194,304 floats
  float* Qw  = (float*)d_ws;
  float* Kw  = Qw + BHCD;
  float* Vw  = Kw + BHCD;
  float* Hid = Vw + BHCD;  // total ws: 4 * 16.8 MB = 67 MB

  qkv_kernel<<<dim3(BC_ / 16, (3 * HD_) / (8 * 64)), 256, 0, stream>>>(
      x, Wq, Wk, Wv, Qw, Kw, Vw);
  rope_kernel<<<dim3((B_ * H_ * C_ * (D_ / 2)) / 256, 2), 256, 0, stream>>>(
      Qw, Kw);
  attn_kernel<<<dim3(C_ / 16 / 8, B_ * H_), 256, 0, stream>>>(Qw, Kw, Vw, Hid);
  proj_kernel<<<dim3(BC_ / 16, HD_ / (8 * 64)), 256, 0, stream>>>(Hid, Wo, out);
}